// Cube2Equi_43593918054826
// MI455X (gfx1250) — compile-verified
//
#include <hip/hip_runtime.h>
#include <math.h>

// Cube (B,6,C,W,W) -> equirect (B,C,2W,4W), bilinear, align_corners=True.
// W=256, C=32, B=2. Output 512x1024 per (b,c).
//
// Memory-bound gather kernel. Per-pixel geometry (face select + bilinear
// weights) is computed ONCE per thread and reused across all 64 (b,c) images.
// Lanes map to consecutive output x -> coalesced gathers & stores (wave32).
// Output is streamed with non-temporal stores so the 100 MB input gather
// footprint stays resident in the 192 MB L2.

#define FACE_W   256
#define OUT_H    512
#define OUT_W    1024
#define CHANNELS 32
#define NBATCH   2
#define FACE_PIX (FACE_W * FACE_W)        // 65536
#define EPSF     1e-9f
#define PI_F     3.14159265358979323846f

__global__ __launch_bounds__(256) void cube2equi_kernel(
    const float* __restrict__ in,   // (2, 6, 32, 256, 256)
    float* __restrict__ out)        // (2, 32, 512, 1024)
{
    const int pix = blockIdx.x * blockDim.x + threadIdx.x;
    if (pix >= OUT_H * OUT_W) return;

    const int ox = pix & (OUT_W - 1);
    const int oy = pix >> 10;

    // ---- per-pixel geometry (amortized over 64 (b,c) images) ----
    const float theta = (2.0f * ((float)ox + 0.5f) / (float)OUT_W - 1.0f) * PI_F;
    const float phi   = (2.0f * ((float)oy + 0.5f) / (float)OUT_H - 1.0f) * (0.5f * PI_F);

    float sth, cth, sph, cph;
    __sincosf(theta, &sth, &cth);     // v_sin_f32 / v_cos_f32 (trans)
    __sincosf(phi,   &sph, &cph);

    const float X = cph * cth;
    const float Y = cph * sth;
    const float Z = sph;

    const float ax = fabsf(X), ay = fabsf(Y), az = fabsf(Z);

    int face;
    if (ax >= ay && ax >= az)      face = (X >= 0.0f) ? 0 : 1;
    else if (ay >= az)             face = (Y >= 0.0f) ? 2 : 3;
    else                           face = (Z >= 0.0f) ? 4 : 5;

    const float axs = fmaxf(ax, EPSF);
    const float ays = fmaxf(ay, EPSF);
    const float azs = fmaxf(az, EPSF);

    float u, v;
    switch (face) {
        case 0:  u =  Y / axs; v =  Z / axs; break;
        case 1:  u = -Y / axs; v =  Z / axs; break;
        case 2:  u = -X / ays; v =  Z / ays; break;
        case 3:  u =  X / ays; v =  Z / ays; break;
        case 4:  u =  Y / azs; v = -X / azs; break;
        default: u =  Y / azs; v =  X / azs; break;
    }

    // align_corners=True: [-1,1] -> [0, W-1]
    const float px = (u + 1.0f) * 0.5f * (float)(FACE_W - 1);
    const float py = (v + 1.0f) * 0.5f * (float)(FACE_W - 1);

    const float x0f = floorf(px);
    const float y0f = floorf(py);
    const float wx  = px - x0f;
    const float wy  = py - y0f;

    const int x0 = min(max((int)x0f,     0), FACE_W - 1);
    const int x1 = min(max((int)x0f + 1, 0), FACE_W - 1);
    const int y0 = min(max((int)y0f,     0), FACE_W - 1);
    const int y1 = min(max((int)y0f + 1, 0), FACE_W - 1);

    const int o00 = y0 * FACE_W + x0;
    const int o01 = y0 * FACE_W + x1;
    const int o10 = y1 * FACE_W + x0;
    const int o11 = y1 * FACE_W + x1;

    const float w00 = (1.0f - wx) * (1.0f - wy);
    const float w01 = wx * (1.0f - wy);
    const float w10 = (1.0f - wx) * wy;
    const float w11 = wx * wy;

    // ---- sweep all (batch, channel) images with the shared footprint ----
    #pragma unroll
    for (int b = 0; b < NBATCH; ++b) {
        const float* __restrict__ src =
            in + ((size_t)(b * 6 + face) * CHANNELS) * FACE_PIX;
        float* __restrict__ dst =
            out + (((size_t)b * CHANNELS) * OUT_H + oy) * OUT_W + ox;

        #pragma unroll 8
        for (int c = 0; c < CHANNELS; ++c) {
            const float* __restrict__ p = src + (size_t)c * FACE_PIX;
            const float v00 = p[o00];
            const float v01 = p[o01];
            const float v10 = p[o10];
            const float v11 = p[o11];
            const float r = fmaf(w00, v00,
                            fmaf(w01, v01,
                            fmaf(w10, v10, w11 * v11)));
            // Write-once streaming output: TH=NT keeps 134 MB of stores out
            // of L2 so the gather footprint (100 MB) stays L2-resident.
            __builtin_nontemporal_store(r, dst + (size_t)c * (OUT_H * OUT_W));
        }
    }
}

extern "C" void kernel_launch(void* const* d_in, const int* in_sizes, int n_in,
                              void* d_out, int out_size, void* d_ws, size_t ws_size,
                              hipStream_t stream) {
    (void)in_sizes; (void)n_in; (void)out_size; (void)d_ws; (void)ws_size;
    const float* in = (const float*)d_in[0];
    float* out = (float*)d_out;

    const int npix = OUT_H * OUT_W;          // 524288 pixels
    const int block = 256;                   // 8 wave32 waves per block
    const int grid = (npix + block - 1) / block;
    cube2equi_kernel<<<grid, block, 0, stream>>>(in, out);
}